// FB8NLLLoss_30923764531209
// MI455X (gfx1250) — compile-verified
//
#include <hip/hip_runtime.h>
#include <math.h>

typedef __attribute__((ext_vector_type(2))) float v2f;
typedef __attribute__((ext_vector_type(8))) float v8f;

#define NTH 128
#define NPH 256
#define NP  (NTH * NPH)

// ---------------------------------------------------------------------------
// Phase 1: Gauss-Legendre nodes/weights (order 128) via fp64 Newton on P_128.
// Matches numpy leggauss to fp64 machine precision; cast to fp32 afterwards.
// ---------------------------------------------------------------------------
__global__ void fb8_gauss_kernel(float* __restrict__ uu, float* __restrict__ ss,
                                 float* __restrict__ lw) {
  int i = threadIdx.x;
  if (i >= NTH) return;
  const int n = NTH;
  double x = cos(3.14159265358979323846 * ((double)i + 0.75) / ((double)n + 0.5));
  double dp = 1.0;
  for (int iter = 0; iter < 12; ++iter) {
    double a = 1.0, b = x;
    for (int k = 2; k <= n; ++k) {
      double c = ((2.0 * k - 1.0) * x * b - (k - 1.0) * a) / (double)k;
      a = b; b = c;
    }
    dp = (double)n * (x * b - a) / (x * x - 1.0);
    x = x - b / dp;
  }
  { // final derivative at converged root for the weight
    double a = 1.0, b = x;
    for (int k = 2; k <= n; ++k) {
      double c = ((2.0 * k - 1.0) * x * b - (k - 1.0) * a) / (double)k;
      a = b; b = c;
    }
    dp = (double)n * (x * b - a) / (x * x - 1.0);
  }
  double w = 2.0 / ((1.0 - x * x) * dp * dp);
  float uf = (float)x;
  float sf = sqrtf(fmaxf(1.0f - uf * uf, 0.0f));           // reference: fp32 sqrt(max(1-u*u,0))
  const float dphi = (float)(2.0 * 3.14159265358979323846 / (double)NPH);
  uu[i] = uf;
  ss[i] = sf;
  lw[i] = logf((float)w) + logf(dphi);                      // reference: fp32 log(w)+log(dphi)
}

// ---------------------------------------------------------------------------
// Phase 2: feature matrix F[NP][8] = {u, s*cos, s*sin, (s*cos)^2, (s*sin)^2,
//                                     logw, 1, 0} per grid point.
// ---------------------------------------------------------------------------
__global__ void fb8_feat_kernel(const float* __restrict__ uu, const float* __restrict__ ss,
                                const float* __restrict__ lw, float* __restrict__ F) {
  int idx = blockIdx.x * blockDim.x + threadIdx.x;
  if (idx >= NP) return;
  int t = idx >> 8;           // theta index (NPH == 256)
  int p = idx & (NPH - 1);    // phi index
  const float dphi = (float)(2.0 * 3.14159265358979323846 / (double)NPH);
  float ph = (float)p * dphi;
  float s  = ss[t];
  float f1 = s * cosf(ph);
  float f2 = s * sinf(ph);
  float* o = F + idx * 8;
  o[0] = uu[t];
  o[1] = f1;
  o[2] = f2;
  o[3] = f1 * f1;
  o[4] = f2 * f2;
  o[5] = lw[t];
  o[6] = 1.0f;
  o[7] = 0.0f;
}

// ---------------------------------------------------------------------------
// Phase 3: main kernel. One block = 16 batch rows. 8 waves stride over the
// 2048 column chunks of 16 grid points; each chunk is two chained
// V_WMMA_F32_16X16X4_F32 ops (K=8), then exp + accumulate per lane.
// ---------------------------------------------------------------------------
__global__ __launch_bounds__(256) void fb8_main_kernel(
    const float* __restrict__ x, const float* __restrict__ theta,
    const float* __restrict__ phi, const float* __restrict__ psi,
    const float* __restrict__ kappa, const float* __restrict__ beta,
    const float* __restrict__ eta, const float* __restrict__ alpha,
    const float* __restrict__ rho, const float* __restrict__ F,
    float* __restrict__ out) {
  __shared__ float partial[8][16];

  const int r0   = blockIdx.x * 16;
  const int lane = threadIdx.x & 31;
  const int wave = threadIdx.x >> 5;
  const int hi   = lane >> 4;     // 0: K=0,1 pair; 1: K=2,3 pair
  const int lo   = lane & 15;

  // ---- A fragment (coefficients), §7.12.2 16x4 fp32 layout ----
  {
  }
  const int rA = r0 + lo;
  const float k_ = kappa[rA];
  const float b_ = beta[rA];
  const float e_ = eta[rA];
  const float al = alpha[rA];
  const float rh = rho[rA];
  const float sa = sinf(al), ca = cosf(al), cr = cosf(rh), sr = sinf(rh);
  // coefficient order c0..c7 = {k*nu0, k*nu1, k*nu2, b, -b*e, 1, -(k+b), 0}
  v2f a0, a1;
  if (hi == 0) {
    a0 = (v2f){k_ * ca, k_ * sa * cr};        // K = 0,1
    a1 = (v2f){-b_ * e_, 1.0f};               // K = 4,5
  } else {
    a0 = (v2f){k_ * sa * sr, b_};             // K = 2,3
    a1 = (v2f){-(k_ + b_), 0.0f};             // K = 6,7
  }

  float acc[8];
#pragma unroll
  for (int i = 0; i < 8; ++i) acc[i] = 0.0f;

  const int chunks = NP / 16;                  // 2048
  for (int c = wave; c < chunks; c += 8) {
    const float* colp = F + ((c * 16 + lo) * 8) + hi * 2;
    v2f b0 = (v2f){colp[0], colp[1]};          // features {0,1} or {2,3}
    v2f b1 = (v2f){colp[4], colp[5]};          // features {4,5} or {6,7}
    v8f d = {};
    d = __builtin_amdgcn_wmma_f32_16x16x4_f32(false, a0, false, b0, (short)0, d,
                                              false, false);
    d = __builtin_amdgcn_wmma_f32_16x16x4_f32(false, a1, false, b1, (short)0, d,
                                              false, false);
#pragma unroll
    for (int i = 0; i < 8; ++i) acc[i] += __expf(d[i]);
  }

  // ---- reduce columns: xor-shuffle within each 16-lane half ----
#pragma unroll
  for (int i = 0; i < 8; ++i) {
    float v = acc[i];
    v += __shfl_xor(v, 1, 32);
    v += __shfl_xor(v, 2, 32);
    v += __shfl_xor(v, 4, 32);
    v += __shfl_xor(v, 8, 32);
    acc[i] = v;
  }
  if (lo == 0) {
#pragma unroll
    for (int i = 0; i < 8; ++i) partial[wave][hi * 8 + i] = acc[i];  // row = hi*8+i
  }
  __syncthreads();

  // ---- finalize: 16 threads, one batch row each ----
  if (threadIdx.x < 16) {
    const int r = r0 + (int)threadIdx.x;
    float S = 0.0f;
#pragma unroll
    for (int w2 = 0; w2 < 8; ++w2) S += partial[w2][threadIdx.x];
    const float kk = kappa[r], bb = beta[r], ee = eta[r];
    const float logC = (kk + bb) + logf(S);   // shift M = kappa + beta added back

    const float th = theta[r], ph2 = phi[r], ps = psi[r];
    const float ct = cosf(th), st = sinf(th);
    const float cp = cosf(ph2), sp = sinf(ph2);
    const float cq = cosf(ps), sq = sinf(ps);
    const float x0 = x[r * 3 + 0], x1 = x[r * 3 + 1], x2 = x[r * 3 + 2];
    // Gamma = H @ K ; g_i x = x . Gamma[:, i]
    const float g1x = x0 * ct + x1 * st * cp + x2 * st * sp;
    const float h1x = -x0 * st + x1 * ct * cp + x2 * ct * sp; // x . H[:,1]
    const float h2x = -x1 * sp + x2 * cp;                     // x . H[:,2]
    const float g2x = h1x * cq + h2x * sq;
    const float g3x = -h1x * sq + h2x * cq;
    const float aa = alpha[r], rr = rho[r];
    const float nu0 = cosf(aa), nu1 = sinf(aa) * cosf(rr), nu2 = sinf(aa) * sinf(rr);
    const float ngx = nu0 * g1x + nu1 * g2x + nu2 * g3x;
    const float logp = kk * ngx + bb * (g2x * g2x - ee * g3x * g3x) - logC;
    out[r] = -logp;
  }
}

// ---------------------------------------------------------------------------
extern "C" void kernel_launch(void* const* d_in, const int* in_sizes, int n_in,
                              void* d_out, int out_size, void* d_ws, size_t ws_size,
                              hipStream_t stream) {
  (void)in_sizes; (void)n_in; (void)ws_size;
  const float* x     = (const float*)d_in[0];
  const float* theta = (const float*)d_in[1];
  const float* phi   = (const float*)d_in[2];
  const float* psi   = (const float*)d_in[3];
  const float* kappa = (const float*)d_in[4];
  const float* beta  = (const float*)d_in[5];
  const float* eta   = (const float*)d_in[6];
  const float* alpha = (const float*)d_in[7];
  const float* rho   = (const float*)d_in[8];
  float* out = (float*)d_out;
  const int B = out_size;                    // 4096

  float* ws = (float*)d_ws;
  float* uu = ws;                            // 128 floats
  float* ss = ws + 128;                      // 128 floats
  float* lw = ws + 256;                      // 128 floats
  float* F  = ws + 512;                      // NP*8 floats = 1 MB

  fb8_gauss_kernel<<<1, 128, 0, stream>>>(uu, ss, lw);
  fb8_feat_kernel<<<NP / 256, 256, 0, stream>>>(uu, ss, lw, F);
  fb8_main_kernel<<<B / 16, 256, 0, stream>>>(x, theta, phi, psi, kappa, beta,
                                              eta, alpha, rho, F, out);
}